// From_ToGCN_67276367725045
// MI455X (gfx1250) — compile-verified
//
#include <hip/hip_runtime.h>
#include <math.h>

// The Python harness owns main(), I/O, device malloc/memcpy, and timing.

typedef __attribute__((ext_vector_type(2))) float v2f;
typedef __attribute__((ext_vector_type(8))) float v8f;

#define NF 128  // feature width of every layer (NFEAT == NHID == NOUT == 128)

// ---------------------------------------------------------------- utilities
__global__ void k_zero(float* __restrict__ p, int n) {
  int i = blockIdx.x * blockDim.x + threadIdx.x;
  if (i < n) p[i] = 0.0f;
}

__global__ void k_degree(const int* __restrict__ esrc, const int* __restrict__ edst,
                         float* __restrict__ deg_out, float* __restrict__ deg_in, int ne) {
  int e = blockIdx.x * blockDim.x + threadIdx.x;
  if (e < ne) {
    atomicAdd(&deg_out[esrc[e]], 1.0f);
    atomicAdd(&deg_in[edst[e]], 1.0f);
  }
}

// in-place: d[i] = clip(d[i],1)^-0.5
__global__ void k_invsqrt_clip(float* __restrict__ d, int n) {
  int i = blockIdx.x * blockDim.x + threadIdx.x;
  if (i < n) {
    float v = d[i];
    v = v < 1.0f ? 1.0f : v;
    d[i] = 1.0f / sqrtf(v);
  }
}

// ------------------------------------------------- WMMA f32 GEMM (+optional 2nd operand)
// out[r][c] = rowscale[r] * ( A1[r][:] @ W1v[:, c]  (+ A2[r][:] @ W2v[:, c]) )
// All matrices are [*, 128] row-major f32.  Block = 256 thr = 8 wave32;
// block owns rows [16*bx, 16*bx+16), wave w owns cols [16w, 16w+16).
__global__ void k_gemm_scale(const float* __restrict__ A1, const float* __restrict__ W1v,
                             const float* __restrict__ A2, const float* __restrict__ W2v,
                             const float* __restrict__ rowscale,
                             float* __restrict__ out, int nrows) {
  __shared__ float sA[16][129];            // pitch 129: 129 % 64 == 1 -> conflict-free cols
  const int m0    = blockIdx.x * 16;
  const int lane  = threadIdx.x & 31;
  const int wave  = threadIdx.x >> 5;      // 0..7
  const int cbase = wave * 16;
  const int half  = lane >> 4;             // 0: lanes 0-15, 1: lanes 16-31
  const int l16   = lane & 15;

  v8f acc = {};

  for (int pass = 0; pass < 2; ++pass) {
    const float* A = pass ? A2 : A1;       // uniform per-block branch
    const float* W = pass ? W2v : W1v;
    if (A == nullptr) break;

    __syncthreads();                       // protect sA reuse across passes
    for (int i = threadIdx.x; i < 16 * NF; i += 256) {
      int r = i >> 7, c = i & (NF - 1);
      int row = m0 + r;
      sA[r][c] = (row < nrows) ? A[row * NF + c] : 0.0f;
    }
    __syncthreads();

#pragma unroll 4
    for (int k = 0; k < NF; k += 4) {
      // A fragment (16x4 f32): lanes 0-15 hold M=l16 K=k,k+1; lanes 16-31 K=k+2,k+3
      const int ka = k + half * 2;
      v2f a, b;
      a.x = sA[l16][ka];
      a.y = sA[l16][ka + 1];
      // B fragment (4x16 f32), mirrored layout: lane holds col cbase+l16
      b.x = W[(ka + 0) * NF + cbase + l16];
      b.y = W[(ka + 1) * NF + cbase + l16];
      acc = __builtin_amdgcn_wmma_f32_16x16x4_f32(
          /*neg_a=*/false, a, /*neg_b=*/false, b,
          /*c_mod=*/(short)0, acc, /*reuse_a=*/false, /*reuse_b=*/false);
    }
  }

  // C/D layout: VGPR v -> row m0+v (lanes 0-15) / m0+v+8 (lanes 16-31), col cbase+l16
#pragma unroll
  for (int v = 0; v < 8; ++v) {
    int row = m0 + v + half * 8;
    if (row < nrows) out[row * NF + cbase + l16] = acc[v] * rowscale[row];
  }
}

// ------------------------------------------------- edge scatter: agg[dst] += S[src]
// one wave per edge, one float4 (4 channels) per lane
__global__ void k_edge_scatter(const float* __restrict__ S, const int* __restrict__ esrc,
                               const int* __restrict__ edst, float* __restrict__ agg, int ne) {
  int e    = (int)((blockIdx.x * (unsigned)blockDim.x + threadIdx.x) >> 5);
  int lane = threadIdx.x & 31;
  if (e >= ne) return;                               // uniform per-wave
  int s = esrc[e], d = edst[e];
  float4 v = ((const float4*)(S + (long long)s * NF))[lane];
  float* drow = agg + (long long)d * NF + lane * 4;
  atomicAdd(drow + 0, v.x);
  atomicAdd(drow + 1, v.y);
  atomicAdd(drow + 2, v.z);
  atomicAdd(drow + 3, v.w);
}

// ------------------------------------------------- H = relu(agg * norm_dst + bias), in-place OK
__global__ void k_bias_relu(const float* __restrict__ agg, const float* __restrict__ norm_dst,
                            const float* __restrict__ bias, float* __restrict__ H, int n) {
  int i = blockIdx.x * blockDim.x + threadIdx.x;
  if (i < n * NF) {
    int node = i >> 7, c = i & (NF - 1);
    float v = agg[i] * norm_dst[node] + bias[c];
    H[i] = v > 0.0f ? v : 0.0f;
  }
}

__global__ void k_count(const int* __restrict__ gid, float* __restrict__ cnt, int n) {
  int i = blockIdx.x * blockDim.x + threadIdx.x;
  if (i < n) atomicAdd(&cnt[gid[i]], 1.0f);
}

// ------------------------------------------------- segment pool (graph_id is sorted)
// 128 threads (= channels), 64 nodes per block, flush one atomic per graph transition
__global__ void k_pool(const float* __restrict__ H, const int* __restrict__ gid,
                       float* __restrict__ out, int n, int colbase) {
  int c    = threadIdx.x;
  int n0   = blockIdx.x * 64;
  int nend = n0 + 64;
  if (nend > n) nend = n;
  float sum = 0.0f;
  int curg = gid[n0];
  for (int node = n0; node < nend; ++node) {
    int g = gid[node];
    if (g != curg) {
      atomicAdd(&out[curg * 256 + colbase + c], sum);
      sum  = 0.0f;
      curg = g;
    }
    sum += H[(long long)node * NF + c];
  }
  atomicAdd(&out[curg * 256 + colbase + c], sum);
}

__global__ void k_div_cnt(float* __restrict__ out, const float* __restrict__ cnt, int total) {
  int i = blockIdx.x * blockDim.x + threadIdx.x;
  if (i < total) {
    float cv = cnt[i >> 8];
    cv = cv < 1.0f ? 1.0f : cv;
    out[i] /= cv;
  }
}

// ---------------------------------------------------------------- driver
extern "C" void kernel_launch(void* const* d_in, const int* in_sizes, int n_in,
                              void* d_out, int out_size, void* d_ws, size_t ws_size,
                              hipStream_t stream) {
  const float* x   = (const float*)d_in[0];   // [N,128]
  const float* W1  = (const float*)d_in[1];   // [128,128]
  const float* b1  = (const float*)d_in[2];   // [128]
  const float* W2  = (const float*)d_in[3];   // [256,128]
  const float* b2  = (const float*)d_in[4];   // [128]
  const int* esrc  = (const int*)d_in[5];     // [E]
  const int* edst  = (const int*)d_in[6];     // [E]
  const int* gid   = (const int*)d_in[7];     // [N] sorted
  float* out       = (float*)d_out;           // [64,256]

  const int N = in_sizes[0] / NF;
  const int E = in_sizes[5];
  const int NGRAPH = out_size / 256;          // 64

  // workspace layout (floats)
  float* ws       = (float*)d_ws;
  float* norm_src = ws;                        // N   (deg_out -> deg_out^-1/2)
  float* norm_dst = ws + N;                    // N   (deg_in  -> deg_in^-1/2)
  float* cnt      = ws + 2 * N;                // NGRAPH
  long long base  = ((2LL * N + NGRAPH + 255) / 256) * 256;
  float* buf1 = ws + base;                     // N*128  scaled projections S1 / S2
  float* buf2 = buf1 + (long long)N * NF;      // N*128  AGG1 -> H1
  float* buf3 = buf2 + (long long)N * NF;      // N*128  AGG2 -> H2
  (void)ws_size; (void)n_in;

  const int T = 256;
  // 1) zero accumulators: norms+cnt, AGG1+AGG2 (contiguous), output
  {
    int nz = (int)base;
    k_zero<<<(nz + T - 1) / T, T, 0, stream>>>(ws, nz);
    int na = 2 * N * NF;
    k_zero<<<(na + T - 1) / T, T, 0, stream>>>(buf2, na);
    k_zero<<<(out_size + T - 1) / T, T, 0, stream>>>(out, out_size);
  }
  // 2) degrees -> norms
  k_degree<<<(E + T - 1) / T, T, 0, stream>>>(esrc, edst, norm_src, norm_dst, E);
  k_invsqrt_clip<<<(2 * N + T - 1) / T, T, 0, stream>>>(ws, 2 * N);

  const int mblocks = (N + 15) / 16;
  const int eblocks = (E + 7) / 8;             // 8 edges (waves) per 256-thr block

  // 3) S1 = (x @ W1) * norm_src      [WMMA]
  k_gemm_scale<<<mblocks, T, 0, stream>>>(x, W1, nullptr, nullptr, norm_src, buf1, N);
  // 4) AGG1[dst] += S1[src]
  k_edge_scatter<<<eblocks, T, 0, stream>>>(buf1, esrc, edst, buf2, E);
  // 5) H1 = relu(AGG1 * norm_dst + b1)   (in place in buf2)
  k_bias_relu<<<(N * NF + T - 1) / T, T, 0, stream>>>(buf2, norm_dst, b1, buf2, N);
  // 6) S2 = (H1 @ W2[0:128] + x @ W2[128:256]) * norm_src   [WMMA, 2 passes]
  k_gemm_scale<<<mblocks, T, 0, stream>>>(buf2, W2, x, W2 + NF * NF, norm_src, buf1, N);
  // 7) AGG2[dst] += S2[src]
  k_edge_scatter<<<eblocks, T, 0, stream>>>(buf1, esrc, edst, buf3, E);
  // 8) H2 = relu(AGG2 * norm_dst + b2)
  k_bias_relu<<<(N * NF + T - 1) / T, T, 0, stream>>>(buf3, norm_dst, b2, buf3, N);
  // 9) per-graph mean pool of [H1 | H2]
  k_count<<<(N + T - 1) / T, T, 0, stream>>>(gid, cnt, N);
  int pblocks = (N + 63) / 64;
  k_pool<<<pblocks, NF, 0, stream>>>(buf2, gid, out, N, 0);
  k_pool<<<pblocks, NF, 0, stream>>>(buf3, gid, out, N, NF);
  k_div_cnt<<<(out_size + T - 1) / T, T, 0, stream>>>(out, cnt, out_size);
}